// Net_6451040878694
// MI455X (gfx1250) — compile-verified
//
#include <hip/hip_runtime.h>
#include <hip/hip_bf16.h>
#include <stddef.h>

#define N_NODES 50000
#define D 128
#define R 8
#define E_EDGES 600000
#define K1 (R * D)          // 1024 : columns of S / rows of flattened W
#define M_TILES (N_NODES / 16)  // 3125, exact

typedef __attribute__((ext_vector_type(2))) float v2f;
typedef __attribute__((ext_vector_type(8))) float v8f;

// ---------------------------------------------------------------------------
// Pass 1: per-(dst, relation) in-degree counts.
// ---------------------------------------------------------------------------
__global__ void rgcn_count_kernel(const int* __restrict__ edge_index,
                                  const int* __restrict__ edge_type,
                                  int* __restrict__ cnt) {
    int e = blockIdx.x * 256 + threadIdx.x;
    if (e < E_EDGES) {
        int dst = edge_index[E_EDGES + e];
        int rel = edge_type[e];
        atomicAdd(&cnt[dst * R + rel], 1);
    }
}

// ---------------------------------------------------------------------------
// Pass 2: scatter pre-scaled source features into S[dst, rel*D + d].
// One wave32 per edge; each lane handles 4 consecutive floats of the row.
// ---------------------------------------------------------------------------
__global__ void __launch_bounds__(256)
rgcn_scatter_kernel(const float* __restrict__ x,
                    const int* __restrict__ edge_index,
                    const int* __restrict__ edge_type,
                    const int* __restrict__ cnt,
                    float* __restrict__ S) {
    int e = blockIdx.x * 8 + (threadIdx.x >> 5);   // 8 waves per block
    if (e >= E_EDGES) return;                      // wave-uniform exit
    int lane = threadIdx.x & 31;

    int src = edge_index[e];
    int dst = edge_index[E_EDGES + e];
    int rel = edge_type[e];

    float c   = (float)cnt[dst * R + rel];
    float inv = 1.0f / fmaxf(c, 1.0f);

    const float4 xv = *(const float4*)(x + (size_t)src * D + lane * 4);
    float* sp = S + (size_t)dst * K1 + rel * D + lane * 4;
    unsafeAtomicAdd(sp + 0, xv.x * inv);
    unsafeAtomicAdd(sp + 1, xv.y * inv);
    unsafeAtomicAdd(sp + 2, xv.z * inv);
    unsafeAtomicAdd(sp + 3, xv.w * inv);
}

// ---------------------------------------------------------------------------
// Pass 3: out = relu( S @ W_flat + x @ W_root + b )
// One wave per 16-row M-tile; wave keeps all 8 column tiles (full 128 cols)
// in registers so S is streamed exactly once. f32 WMMA 16x16x4.
//
// A-operand layout (32-bit A 16x4): lanes 0-15 -> rows M=0..15, VGPR0=K0,
// VGPR1=K1; lanes 16-31 -> same rows, VGPR0=K2, VGPR1=K3. So lane holds the
// float2 at (m0 + lane%16, k + 2*(lane/16)).
// B-operand (4x16 row-striped): VGPR j, lanes 0-15 = row K=(k+j), lanes
// 16-31 = row K=(k+2+j), cols = lane%16.
// C layout: VGPR i -> (M = i + 8*(lane/16), N = lane%16).
// ---------------------------------------------------------------------------
__global__ void __launch_bounds__(128)
rgcn_gemm_kernel(const float* __restrict__ S,
                 const float* __restrict__ x,
                 const float* __restrict__ W,       // [R*D, D] flattened
                 const float* __restrict__ Wroot,   // [D, D]
                 const float* __restrict__ bias,    // [D]
                 float* __restrict__ out) {
    int wave = blockIdx.x * 4 + (threadIdx.x >> 5);
    if (wave >= M_TILES) return;                   // wave-uniform exit
    int lane = threadIdx.x & 31;
    int lr = lane & 15;
    int hi = lane >> 4;
    int m0 = wave * 16;

    v8f acc[8] = {};

    // ---- main K loop over aggregated features: S @ W_flat -----------------
    {
        const float* arow = S + (size_t)(m0 + lr) * K1 + 2 * hi;
        for (int k = 0; k < K1; k += 4) {
            v2f a = *(const v2f*)(arow + k);
            int kk = k + 2 * hi;
            const float* wk0 = W + (size_t)kk * D;
#pragma unroll
            for (int t = 0; t < 8; ++t) {
                int n = t * 16 + lr;
                v2f bb;
                bb.x = wk0[n];
                bb.y = wk0[D + n];
                acc[t] = __builtin_amdgcn_wmma_f32_16x16x4_f32(
                    false, a, false, bb, (short)0, acc[t], false, false);
            }
        }
    }

    // ---- root term: x @ W_root -------------------------------------------
    {
        const float* arow = x + (size_t)(m0 + lr) * D + 2 * hi;
        for (int k = 0; k < D; k += 4) {
            v2f a = *(const v2f*)(arow + k);
            int kk = k + 2 * hi;
            const float* wk0 = Wroot + (size_t)kk * D;
#pragma unroll
            for (int t = 0; t < 8; ++t) {
                int n = t * 16 + lr;
                v2f bb;
                bb.x = wk0[n];
                bb.y = wk0[D + n];
                acc[t] = __builtin_amdgcn_wmma_f32_16x16x4_f32(
                    false, a, false, bb, (short)0, acc[t], false, false);
            }
        }
    }

    // ---- epilogue: bias + relu, store ------------------------------------
#pragma unroll
    for (int t = 0; t < 8; ++t) {
        int n = t * 16 + lr;
        float bt = bias[n];
#pragma unroll
        for (int i = 0; i < 8; ++i) {
            float v = acc[t][i] + bt;
            out[(size_t)(m0 + i + 8 * hi) * D + n] = v > 0.0f ? v : 0.0f;
        }
    }
}

// ---------------------------------------------------------------------------
extern "C" void kernel_launch(void* const* d_in, const int* in_sizes, int n_in,
                              void* d_out, int out_size, void* d_ws, size_t ws_size,
                              hipStream_t stream) {
    const float* x   = (const float*)d_in[0];   // [N, D]
    const int*   ei  = (const int*)d_in[1];     // [2, E]
    const int*   et  = (const int*)d_in[2];     // [E]
    const float* W   = (const float*)d_in[3];   // [R, D, D] == [R*D, D] flat
    const float* Wr  = (const float*)d_in[4];   // [D, D]
    const float* b   = (const float*)d_in[5];   // [D]
    float* out = (float*)d_out;

    // Workspace layout: S [N, R*D] f32, then cnt [N*R] int.
    float* S = (float*)d_ws;
    size_t S_bytes = (size_t)N_NODES * K1 * sizeof(float);      // 204.8 MB
    int* cnt = (int*)((char*)d_ws + S_bytes);
    size_t cnt_bytes = (size_t)N_NODES * R * sizeof(int);       // 1.6 MB

    hipMemsetAsync(d_ws, 0, S_bytes + cnt_bytes, stream);

    rgcn_count_kernel<<<(E_EDGES + 255) / 256, 256, 0, stream>>>(ei, et, cnt);

    rgcn_scatter_kernel<<<(E_EDGES + 7) / 8, 256, 0, stream>>>(x, ei, et, cnt, S);

    rgcn_gemm_kernel<<<(M_TILES + 3) / 4, 128, 0, stream>>>(S, x, W, Wr, b, out);
}